// HoloAttentionV2_68272800137363
// MI455X (gfx1250) — compile-verified
//
#include <hip/hip_runtime.h>
#include <math.h>

// ---------------- problem constants ----------------
#define B_   2
#define T_   8192
#define H_   16
#define D_   64
#define HD_  1024
#define DM_  1024
#define M_   (B_ * T_)          // 16384 rows
#define PHASE_SCALE 10.0f
#define LN10000     9.210340371976184f
#define TWO_PI      6.28318530717958647692f

// scan decomposition
#define NCHUNK 32
#define CHUNK  (T_ / NCHUNK)    // 256

typedef __attribute__((ext_vector_type(2))) float v2f;
typedef __attribute__((ext_vector_type(8))) float v8f;

// =====================================================================
// GEMM: C[M,N] = X[M,K] * W[N,K]^T  (all fp32, V_WMMA_F32_16X16X4_F32)
// Block tile 128x128, K-chunk 16, 8 waves; each wave owns a 64x32 region
// (4x2 tiles of 16x16).
//
// LDS layout: k-PAIR major, v2f elements:  Xs[kpair][row] = {X[2kp][row],
// X[2kp+1][row]}.  A lane's WMMA fragment (K = kk+khalf, kk+khalf+1) is
// then ONE aligned ds_load_b64 directly into the even VGPR pair the WMMA
// reads - no operand repacking movs.  Double-buffered so the next K-chunk
// streams from global while the current one feeds the matrix pipe.
// =====================================================================
#define TM   128
#define TN   128
#define KC   16
#define KP   (KC / 2)   // 8 pair-planes
#define LDP  132        // pair-plane stride (v2f units), padded

__device__ __forceinline__
void stage_tiles(v2f (*__restrict__ Xs)[LDP], v2f (*__restrict__ Ws)[LDP],
                 const float* __restrict__ X, const float* __restrict__ W,
                 int K, int m0, int n0, int k0, int sr, int sk8)
{
    const int kp0 = sk8 >> 1;   // 0 or 4
    const float4 x0 = *(const float4*)(X + (size_t)(m0 + sr) * K + k0 + sk8);
    const float4 x1 = *(const float4*)(X + (size_t)(m0 + sr) * K + k0 + sk8 + 4);
    Xs[kp0 + 0][sr] = v2f{x0.x, x0.y};
    Xs[kp0 + 1][sr] = v2f{x0.z, x0.w};
    Xs[kp0 + 2][sr] = v2f{x1.x, x1.y};
    Xs[kp0 + 3][sr] = v2f{x1.z, x1.w};

    const float4 w0 = *(const float4*)(W + (size_t)(n0 + sr) * K + k0 + sk8);
    const float4 w1 = *(const float4*)(W + (size_t)(n0 + sr) * K + k0 + sk8 + 4);
    Ws[kp0 + 0][sr] = v2f{w0.x, w0.y};
    Ws[kp0 + 1][sr] = v2f{w0.z, w0.w};
    Ws[kp0 + 2][sr] = v2f{w1.x, w1.y};
    Ws[kp0 + 3][sr] = v2f{w1.z, w1.w};
}

__global__ __launch_bounds__(256)
void gemm_xwt(const float* __restrict__ X, const float* __restrict__ W,
              float* __restrict__ C, int M, int N, int K)
{
    __shared__ v2f Xs[2][KP][LDP];
    __shared__ v2f Ws[2][KP][LDP];

    const int nBN  = N / TN;
    const int bm   = blockIdx.x / nBN;
    const int bn   = blockIdx.x % nBN;
    const int tid  = threadIdx.x;
    const int lane = tid & 31;
    const int wave = tid >> 5;
    const int wr   = wave >> 2;            // 0..1 : row group (64 rows)
    const int wc   = wave & 3;             // 0..3 : col group (32 cols)
    const int l16  = lane & 15;
    const int kq   = lane >> 4;            // 0/1 : which K pair of the 4

    const v8f vzero = {};
    v8f acc[4][2];
#pragma unroll
    for (int i = 0; i < 4; ++i)
#pragma unroll
        for (int j = 0; j < 2; ++j)
            acc[i][j] = vzero;

    const int m0 = bm * TM;
    const int n0 = bn * TN;

    // staging assignment: 2 threads per row, 8 contiguous floats each
    const int sr  = tid >> 1;              // 0..127
    const int sk8 = (tid & 1) << 3;        // 0 or 8

    // prologue: stage first K-chunk into buffer 0
    stage_tiles(Xs[0], Ws[0], X, W, K, m0, n0, 0, sr, sk8);
    __syncthreads();

    for (int k0 = 0; k0 < K; k0 += KC) {
        const int buf  = (k0 / KC) & 1;
        const int nbuf = buf ^ 1;

        // stream next chunk into the other buffer while computing this one
        if (k0 + KC < K) {
            stage_tiles(Xs[nbuf], Ws[nbuf], X, W, K, m0, n0, k0 + KC, sr, sk8);
            __builtin_prefetch(X + (size_t)(m0 + sr) * K + k0 + 2 * KC + sk8, 0, 0);
            __builtin_prefetch(W + (size_t)(n0 + sr) * K + k0 + 2 * KC + sk8, 0, 0);
        }

        // ---- 4 WMMA K-steps of 4 over this K-chunk ----
#pragma unroll
        for (int kk = 0; kk < KC; kk += 4) {
            const int kp = (kk >> 1) + kq;   // pair-plane for this half-wave
            v2f a[4], bf[2];
#pragma unroll
            for (int rt = 0; rt < 4; ++rt)
                a[rt] = Xs[buf][kp][wr * 64 + rt * 16 + l16];
#pragma unroll
            for (int ct = 0; ct < 2; ++ct)
                bf[ct] = Ws[buf][kp][wc * 32 + ct * 16 + l16];
#pragma unroll
            for (int rt = 0; rt < 4; ++rt)
#pragma unroll
                for (int ct = 0; ct < 2; ++ct)
                    acc[rt][ct] = __builtin_amdgcn_wmma_f32_16x16x4_f32(
                        false, a[rt], false, bf[ct],
                        (short)0, acc[rt][ct], false, false);
        }
        __syncthreads();
    }

    // ---- store: C/D layout -> lanes 0-15: M=j, lanes 16-31: M=8+j; N=l16
#pragma unroll
    for (int rt = 0; rt < 4; ++rt)
#pragma unroll
        for (int ct = 0; ct < 2; ++ct) {
            const int rowBase = m0 + wr * 64 + rt * 16 + (kq << 3);
            const int col     = n0 + wc * 32 + ct * 16 + l16;
#pragma unroll
            for (int j = 0; j < 8; ++j)
                C[(size_t)(rowBase + j) * N + col] = acc[rt][ct][j];
        }
}

// =====================================================================
// Scan phase 1: per-(b,h,chunk) partial sums of the 4 scan components.
// 64 threads = D lanes -> coalesced 256B reads of v / k_real rows.
// =====================================================================
__device__ __forceinline__ float chan_freq(int h, int d)
{
    return TWO_PI * ((float)(h + 1) / (float)H_) *
           __expf(-((float)d / (float)D_) * LN10000);
}

__global__ __launch_bounds__(64)
void scan_partial(const float* __restrict__ V, const float* __restrict__ Kr,
                  float4* __restrict__ sums)
{
    const int blk   = blockIdx.x;
    const int chunk = blk % NCHUNK;
    const int bh    = blk / NCHUNK;      // b*H + h
    const int h     = bh % H_;
    const int b     = bh / H_;
    const int d     = threadIdx.x;
    const int col   = h * D_ + d;
    const float freq = chan_freq(h, d);

    float spr = 0.f, spi = 0.f, sar = 0.f, sai = 0.f;
    const int t0 = chunk * CHUNK;
    for (int j = 0; j < CHUNK; ++j) {
        const int t = t0 + j;
        const float v = V[(size_t)(b * T_ + t) * HD_ + col];
        float srot, crot;
        __sincosf((float)t * freq, &srot, &crot);
        spr += v * crot;
        spi += v * srot;
        if (t > 0) {
            const float kp = PHASE_SCALE * Kr[(size_t)(b * T_ + t - 1) * HD_ + col];
            float sk, ck;
            __sincosf(kp, &sk, &ck);
            sar += v * ck;     // Re of v*conj(k_prev)
            sai -= v * sk;     // Im of v*conj(k_prev)
        }
    }
    const int ch = bh * D_ + d;          // channel id 0..2047
    sums[(size_t)ch * NCHUNK + chunk] = make_float4(spr, spi, sar, sai);
}

// =====================================================================
// Scan phase 2: exclusive prefix over the 32 chunk sums of each channel.
// =====================================================================
__global__ void scan_prefix(float4* __restrict__ sums)
{
    const int ch = blockIdx.x * blockDim.x + threadIdx.x;
    if (ch >= B_ * H_ * D_) return;
    float4 run = make_float4(0.f, 0.f, 0.f, 0.f);
    for (int c = 0; c < NCHUNK; ++c) {
        const float4 s = sums[(size_t)ch * NCHUNK + c];
        sums[(size_t)ch * NCHUNK + c] = run;
        run.x += s.x; run.y += s.y; run.z += s.z; run.w += s.w;
    }
}

// =====================================================================
// Scan phase 3: replay chunk with carried offset, emit gated combine.
// =====================================================================
__global__ __launch_bounds__(64)
void scan_final(const float* __restrict__ V, const float* __restrict__ Kr,
                const float4* __restrict__ offs, const float* __restrict__ gate,
                float* __restrict__ Cmb)
{
    const int blk   = blockIdx.x;
    const int chunk = blk % NCHUNK;
    const int bh    = blk / NCHUNK;
    const int h     = bh % H_;
    const int b     = bh / H_;
    const int d     = threadIdx.x;
    const int col   = h * D_ + d;
    const float freq = chan_freq(h, d);
    const int ch    = bh * D_ + d;

    const float4 o = offs[(size_t)ch * NCHUNK + chunk];
    float pr = o.x, pi = o.y, ar = o.z, ai = o.w;
    const float g0 = gate[h * 2 + 0];
    const float g1 = gate[h * 2 + 1];

    const int t0 = chunk * CHUNK;
    for (int j = 0; j < CHUNK; ++j) {
        const int t = t0 + j;
        const float v = V[(size_t)(b * T_ + t) * HD_ + col];
        float srot, crot;
        __sincosf((float)t * freq, &srot, &crot);
        pr += v * crot;
        pi += v * srot;
        if (t > 0) {
            const float kp = PHASE_SCALE * Kr[(size_t)(b * T_ + t - 1) * HD_ + col];
            float sk, ck;
            __sincosf(kp, &sk, &ck);
            ar += v * ck;
            ai -= v * sk;
        }
        const float qa = PHASE_SCALE * Kr[(size_t)(b * T_ + t) * HD_ + col];
        float sq, cq;
        __sincosf(qa, &sq, &cq);
        const float inv = rsqrtf((float)(t + 1));
        const float opos = (pr * crot + pi * srot) * inv;   // Re(mem_pos*conj(rot))
        const float oass = (ar * cq - ai * sq) * inv;       // Re(mem_assoc*q)
        Cmb[(size_t)(b * T_ + t) * HD_ + col] = opos * g0 + oass * g1;
    }
}

// =====================================================================
// launch
// =====================================================================
extern "C" void kernel_launch(void* const* d_in, const int* in_sizes, int n_in,
                              void* d_out, int out_size, void* d_ws, size_t ws_size,
                              hipStream_t stream)
{
    const float* x    = (const float*)d_in[0];
    const float* Wk   = (const float*)d_in[1];
    const float* Wv   = (const float*)d_in[2];
    const float* Wo   = (const float*)d_in[3];
    const float* gate = (const float*)d_in[4];
    float* out = (float*)d_out;

    // workspace layout
    float*  kreal = (float*)d_ws;                          // M_*HD_ = 64 MiB
    float*  vbuf  = kreal + (size_t)M_ * HD_;              // 64 MiB
    float*  cmb   = vbuf  + (size_t)M_ * HD_;              // 64 MiB
    float4* sums  = (float4*)(cmb + (size_t)M_ * HD_);     // 2048*32*16B = 1 MiB

    const dim3 gemm_grid((M_ / TM) * (HD_ / TN));          // 128*8 = 1024 blocks
    const dim3 gemm_blk(256);

    // k_real = x @ Wk^T ; v = x @ Wv^T
    gemm_xwt<<<gemm_grid, gemm_blk, 0, stream>>>(x, Wk, kreal, M_, HD_, DM_);
    gemm_xwt<<<gemm_grid, gemm_blk, 0, stream>>>(x, Wv, vbuf,  M_, HD_, DM_);

    // decoupled 3-phase scan + gated combine
    scan_partial<<<dim3(B_ * H_ * NCHUNK), dim3(64), 0, stream>>>(vbuf, kreal, sums);
    scan_prefix <<<dim3((B_ * H_ * D_ + 255) / 256), dim3(256), 0, stream>>>(sums);
    scan_final  <<<dim3(B_ * H_ * NCHUNK), dim3(64), 0, stream>>>(vbuf, kreal, sums, gate, cmb);

    // out = combined @ Wo^T
    gemm_xwt<<<gemm_grid, gemm_blk, 0, stream>>>(cmb, Wo, out, M_, DM_, HD_);
}